// Likelyloss_51634096833152
// MI455X (gfx1250) — compile-verified
//
#include <hip/hip_runtime.h>

typedef __attribute__((ext_vector_type(2))) float v2f;
typedef __attribute__((ext_vector_type(8))) float v8f;

#define XY 102400            // 320*320 pixels per channel plane
#define NB 32                // batch
#define BPB 10               // blocks per batch image
#define TILE 512             // pixels staged in LDS per tile
#define TILES_PER_BLOCK 20   // 10 blocks * 20 tiles * 512 px = 102400 px per b
#define NTHREADS 256         // 8 waves (wave32)
#define WAVE_PIX (TILE / 8)  // 64 pixels per wave per tile
#define PSTRIDE 16           // floats per pixel group in LDS (64B, conflict-free)
#define EPSV 1e-10f

// COMB[m][k][c]
__constant__ float c_COMB[3][4][4] = {
    {{1,0,0,0},{0,1,1,0},{0,1,1,0},{0,0,0,1}},
    {{1,0,0,0},{0,1,0,0},{0,0,1,1},{0,0,1,1}},
    {{1,0,0,0},{0,1,1,1},{0,1,1,1},{0,1,1,1}}};
// MU_GT[k][m]
__constant__ float c_MUGT[4][3] = {
    {0.7711f,-0.3106f,1.192f},
    {-0.7858f,0.7168f,-0.5759f},
    {-0.6861f,1.5934f,-0.2782f},
    {0.4515f,1.637f,0.1207f}};

// ws layout: [0,384) T[b][c][m], [384,512) U[b][c] (stride 4), [512,517) P[ch]
__global__ void zero_ws_kernel(float* __restrict__ ws) {
  int i = blockIdx.x * blockDim.x + threadIdx.x;
  if (i < 517) ws[i] = 0.0f;
}

// LDS pixel-group slots:
//   0..3  : mask * pred[1+c]   (A rows 0..3)
//   4..8  : pred[0..4]         (A rows 4..8)
//   9..11 : inputs[0..2]       (B cols 0..2)
//   12    : 1.0f               (B col 3 -> U, P sums)
//   13    : 0.0f               (A rows 9..15, B cols 4..15)
__global__ __launch_bounds__(NTHREADS) void reduce_kernel(
    const float* __restrict__ pred, const float* __restrict__ inp,
    const int* __restrict__ heart, float* __restrict__ T,
    float* __restrict__ U, float* __restrict__ P) {
  __shared__ float lds[TILE * PSTRIDE];

  const int b   = blockIdx.x / BPB;
  const int blk = blockIdx.x % BPB;
  const float* pb = pred  + (size_t)b * 5 * XY;
  const float* ib = inp   + (size_t)b * 3 * XY;
  const int*   hb = heart + (size_t)b * XY;

  const int tid  = threadIdx.x;
  const int lane = tid & 31;
  const int wave = tid >> 5;
  const int half = lane >> 4;  // K-slice selector per WMMA f32 A/B layout
  const int idx  = lane & 15;  // A row (M) / B col (N)

  // loop-invariant per-lane LDS slot selection (no branches in hot loop)
  const int slotA = (idx < 9) ? idx : 13;
  const int slotB = (idx < 3) ? (9 + idx) : ((idx == 3) ? 12 : 13);

  v8f acc = {};

  const int pstart = blk * (TILES_PER_BLOCK * TILE);
  for (int t = 0; t < TILES_PER_BLOCK; ++t) {
    const int pix0 = pstart + t * TILE;
    // ---- coalesced global -> LDS stage: 2 pixels per thread ----
    {
      const int g = tid * 2;          // local pixel pair
      const int pg = pix0 + g;        // global pixel index within plane
      float2 pr0 = *(const float2*)(pb + 0 * (size_t)XY + pg);
      float2 pr1 = *(const float2*)(pb + 1 * (size_t)XY + pg);
      float2 pr2 = *(const float2*)(pb + 2 * (size_t)XY + pg);
      float2 pr3 = *(const float2*)(pb + 3 * (size_t)XY + pg);
      float2 pr4 = *(const float2*)(pb + 4 * (size_t)XY + pg);
      float2 in0 = *(const float2*)(ib + 0 * (size_t)XY + pg);
      float2 in1 = *(const float2*)(ib + 1 * (size_t)XY + pg);
      float2 in2 = *(const float2*)(ib + 2 * (size_t)XY + pg);
      int2   h   = *(const int2*)(hb + pg);
      const float m0 = (h.x == 1) ? 1.f : 0.f;
      const float m1 = (h.y == 1) ? 1.f : 0.f;

      float* q0 = &lds[(g + 0) * PSTRIDE];   // 64B aligned
      *(float4*)(q0 + 0)  = make_float4(m0 * pr1.x, m0 * pr2.x, m0 * pr3.x, m0 * pr4.x);
      *(float4*)(q0 + 4)  = make_float4(pr0.x, pr1.x, pr2.x, pr3.x);
      *(float4*)(q0 + 8)  = make_float4(pr4.x, in0.x, in1.x, in2.x);
      *(float4*)(q0 + 12) = make_float4(1.f, 0.f, 0.f, 0.f);

      float* q1 = &lds[(g + 1) * PSTRIDE];
      *(float4*)(q1 + 0)  = make_float4(m1 * pr1.y, m1 * pr2.y, m1 * pr3.y, m1 * pr4.y);
      *(float4*)(q1 + 4)  = make_float4(pr0.y, pr1.y, pr2.y, pr3.y);
      *(float4*)(q1 + 8)  = make_float4(pr4.y, in0.y, in1.y, in2.y);
      *(float4*)(q1 + 12) = make_float4(1.f, 0.f, 0.f, 0.f);
    }
    __syncthreads();

    // ---- v_wmma_f32_16x16x4_f32: K = pixel dim, 4 pixels per issue ----
    // Per iteration: 4 unconditional ds_load_b32 (bank-conflict-free) + 1 WMMA.
    const int wbase = wave * WAVE_PIX;
#pragma unroll 8
    for (int it = 0; it < WAVE_PIX / 4; ++it) {
      const int o0 = (wbase + it * 4 + 2 * half) * PSTRIDE;  // K=0/2 pixel
      const int o1 = o0 + PSTRIDE;                           // K=1/3 pixel
      float a0 = lds[o0 + slotA];
      float a1 = lds[o1 + slotA];
      float b0 = lds[o0 + slotB];
      float b1 = lds[o1 + slotB];
      v2f A = {a0, a1};
      v2f B = {b0, b1};
      acc = __builtin_amdgcn_wmma_f32_16x16x4_f32(
          false, A, false, B, (short)0, acc, false, false);
    }
    __syncthreads();
  }

  // ---- scatter accumulator tile: D[M][N], M = r + 8*half, N = idx ----
  const int Ncol = idx;
#pragma unroll
  for (int r = 0; r < 8; ++r) {
    const int M = r + 8 * half;
    const float v = acc[r];
    if (M < 4) {
      if (Ncol < 3)        atomicAdd(&T[b * 12 + M * 3 + Ncol], v);
      else if (Ncol == 3)  atomicAdd(&U[b * 4 + M], v);
    } else if (M < 9 && Ncol == 3) {
      atomicAdd(&P[M - 4], v);
    }
  }
}

__global__ void finalize_kernel(const float* __restrict__ T,
                                const float* __restrict__ U,
                                const float* __restrict__ P,
                                float* __restrict__ out) {
  __shared__ float red[NB * 12];
  const int t = threadIdx.x;  // 0..383 -> (b,k,m)
  const int b = t / 12;
  const int k = (t % 12) / 3;
  const int m = t % 3;
  float num = 0.f, den = EPSV;
#pragma unroll
  for (int c = 0; c < 4; ++c) {
    const float w = c_COMB[m][k][c];
    num += w * T[b * 12 + c * 3 + m];
    den += w * U[b * 4 + c];
  }
  const float mu = num / den;
  const float d = c_MUGT[k][m] - mu;
  red[t] = d * d;
  __syncthreads();
  if (t == 0) {
    float s = 0.f;
    for (int i = 0; i < NB * 12; ++i) s += red[i];
    const float likelyloss = s / (float)(NB * 12);
    const float invN = 1.0f / (float)((size_t)NB * XY);
    const float a = P[0] * invN;
    const float inv = 1.0f / (1.0f - a);
    float l2 = 0.f;
    for (int j = 1; j < 5; ++j) {
      const float q = P[j] * invN * inv - 0.25f;
      l2 += q * q;
    }
    out[0] = likelyloss + l2;
  }
}

extern "C" void kernel_launch(void* const* d_in, const int* in_sizes, int n_in,
                              void* d_out, int out_size, void* d_ws,
                              size_t ws_size, hipStream_t stream) {
  const float* pred  = (const float*)d_in[0];
  const float* inp   = (const float*)d_in[1];
  const int*   heart = (const int*)d_in[2];
  // d_in[3] (gt) is unused by the reference -> never touched (saves 65 MB of HBM traffic)
  float* ws = (float*)d_ws;
  float* T = ws;         // 384 floats
  float* U = ws + 384;   // 128 floats
  float* P = ws + 512;   // 5 floats

  zero_ws_kernel<<<3, 256, 0, stream>>>(ws);
  reduce_kernel<<<NB * BPB, NTHREADS, 0, stream>>>(pred, inp, heart, T, U, P);
  finalize_kernel<<<1, NB * 12, 0, stream>>>(T, U, P, (float*)d_out);
}